// DeformConv1d_9199819948336
// MI455X (gfx1250) — compile-verified
//
#include <hip/hip_runtime.h>
#include <hip/hip_bf16.h>
#include <math.h>

// Problem constants (from reference)
constexpr int BATCH = 8;
constexpr int CIN   = 256;
constexpr int COUT  = 256;
constexpr int LEN   = 8192;
constexpr int KW    = 5;
constexpr int GRP   = 4;
constexpr int PADW  = 2;
constexpr int LOUT  = 8192;          // stride 1, dil 1, pad 2 -> same length
constexpr int COFF  = CIN * KW * 2;  // 2560
constexpr int CPG   = CIN / GRP;     // 64
constexpr int KD    = CPG * KW;      // 320  (GEMM K-dim per group)
constexpr int OPG   = COFF / GRP;    // 640  (offset-conv out channels per group)
constexpr int LT    = 16;            // L-tile == WMMA N
constexpr int NTHREADS = 512;        // 16 waves (wave32)
constexpr int NWAVES   = NTHREADS / 32;

// LDS tiles (floats)
constexpr int SX_STRIDE = LT + KW - 1;            // 20: raw x tile cols per channel
constexpr int SX_SIZE   = CIN * SX_STRIDE;        // 5120 (transient, aliases sOff)
constexpr int SB_ROWS   = CIN * KW;               // 1280 im2col rows (== val rows)
constexpr int SB_SIZE   = SB_ROWS * LT;           // 20480
constexpr int SOFF_SIZE = COFF * LT;              // 40960
// Region 0: im2col B tile, later reused as the val tile (dead after phase 2).
// Region 1: offset-conv output; its first SX_SIZE floats host the raw x tile
//           during phase 1 (dead before phase 2 writes it).
constexpr size_t SMEM_BYTES =
    (size_t)(SB_SIZE + SOFF_SIZE) * sizeof(float);  // 245,760 B < 320 KB

typedef __attribute__((ext_vector_type(2))) float v2f;
typedef __attribute__((ext_vector_type(8))) float v8f;

__device__ __forceinline__ v8f wmma_f32_16x16x4(v2f a, v2f b, v8f c) {
    // 8 args: (neg_a, A, neg_b, B, c_mod, C, reuse_a, reuse_b)
    return __builtin_amdgcn_wmma_f32_16x16x4_f32(
        false, a, false, b, (short)0, c, false, false);
}

__global__ __launch_bounds__(NTHREADS)
void deform_conv1d_fused(const float* __restrict__ x,
                         const float* __restrict__ w_off,
                         const float* __restrict__ b_off,
                         const float* __restrict__ weight,
                         const float* __restrict__ bias,
                         float* __restrict__ out) {
    extern __shared__ float smem[];
    float* sB   = smem;                      // [1280][16] im2col(x) tile (phase 1-2)
    float* sVal = smem;                      // [1280][16] gathered*attn (phase 3-4), aliases sB
    float* sOff = smem + SB_SIZE;            // [2560][16] offset-conv out
    float* sX   = sOff;                      // [256][20] raw x tile (phase 1 only)

    const int l0   = blockIdx.x * LT;
    const int b    = blockIdx.y;
    const int tid  = threadIdx.x;
    const int wave = tid >> 5;
    const int lane = tid & 31;
    const int nn    = lane & 15;             // WMMA N index / A M-row index
    const int khalf = lane >> 4;             // 0: K{0,1}, 1: K{2,3}

    // ---- Phase 1a: stage raw x tile (with conv zero-padding) into LDS ----
    for (int i = tid; i < SX_SIZE; i += NTHREADS) {
        const int c = i / SX_STRIDE;
        const int j = i - c * SX_STRIDE;
        const int p = l0 - PADW + j;
        sX[i] = (p >= 0 && p < LEN) ? x[((size_t)b * CIN + c) * LEN + p] : 0.0f;
    }
    __syncthreads();

    // ---- Phase 1b: expand to im2col B tile: sB[(c*5+k)*16 + n] = x[b,c,l0+n-2+k]
    // Does the /5 decode ONCE per element instead of per-WMMA in the hot loop.
    for (int i = tid; i < SB_SIZE; i += NTHREADS) {
        const int mg = i >> 4;               // c*5 + k
        const int n  = i & 15;
        const int c  = mg / KW;
        const int k  = mg - c * KW;
        sB[i] = sX[c * SX_STRIDE + n + k];
    }
    __syncthreads();

    // ---- Phase 2: offset conv GEMM -> sOff (2560 x 16) via V_WMMA_F32_16X16X4_F32
    // Per group g: D(640 x 16) = Woff_g(640 x 320) * B_g(320 x 16); B addressing
    // is now linear (stride 64 B per k), so the inner loop is pure load+wmma.
    for (int t = wave; t < COFF / 16; t += NWAVES) {     // 160 M-tiles, 10 per wave
        const int co0 = t * 16;
        const int g   = co0 / OPG;
        v8f acc = {};
        const float* wrow = w_off + (size_t)(co0 + nn) * KD + khalf * 2;
        const float* brow = sB + ((size_t)g * KD + khalf * 2) * LT + nn;
        __builtin_prefetch(wrow, 0, 1);                  // global_prefetch_b8
        for (int ks = 0; ks < KD / 4; ++ks) {            // 80 K-steps
            v2f a = *(const v2f*)(wrow + ks * 4);        // 8B-aligned pair
            v2f bb;
            bb.x = brow[ks * 4 * LT];
            bb.y = brow[ks * 4 * LT + LT];
            acc = wmma_f32_16x16x4(a, bb, acc);
        }
        // D layout: VGPR r -> (M = r + 8*khalf, N = nn)
        #pragma unroll
        for (int r = 0; r < 8; ++r) {
            const int row = co0 + r + khalf * 8;
            sOff[row * LT + nn] = acc[r] + b_off[row];
        }
    }
    __syncthreads();

    // ---- Phase 3: group-softmax on attn, fractional gather from x, weight by attn
    // offset = sOff rows [0,1280), raw attn = rows [1280,2560).
    // softmax couples rows 1280 + {m, m+320, m+640, m+960}, m = mg % 320.
    // sVal overwrites sB (dead after phase 2).
    for (int i = tid; i < SB_SIZE; i += NTHREADS) {      // 20480 elems, 40 per thread
        const int mg = i >> 4;                           // c*5 + k, 0..1279
        const int n  = i & 15;
        const int c  = mg / KW;
        const int k  = mg - c * KW;
        const int gp   = mg / KD;                        // own softmax group
        const int midx = mg - gp * KD;                   // 0..319
        const float r0 = sOff[(COFF / 2 + midx          ) * LT + n];
        const float r1 = sOff[(COFF / 2 + midx +     KD ) * LT + n];
        const float r2 = sOff[(COFF / 2 + midx + 2 * KD ) * LT + n];
        const float r3 = sOff[(COFF / 2 + midx + 3 * KD ) * LT + n];
        const float mx = fmaxf(fmaxf(r0, r1), fmaxf(r2, r3));
        const float e0 = __expf(r0 - mx), e1 = __expf(r1 - mx);
        const float e2 = __expf(r2 - mx), e3 = __expf(r3 - mx);
        const float esum = e0 + e1 + e2 + e3;
        const float eown = (gp == 0) ? e0 : (gp == 1) ? e1 : (gp == 2) ? e2 : e3;
        const float attn = eown * __builtin_amdgcn_rcpf(esum);  // v_rcp_f32

        const float offv = sOff[mg * LT + n];
        const float pos  = (float)(l0 + n - PADW + k) + offv;
        const float i0f  = floorf(pos);
        const float frac = pos - i0f;
        const int   i0   = (int)i0f;
        const int   i1   = i0 + 1;
        const float* xrow = x + ((size_t)b * CIN + c) * LEN;
        const float v0 = (i0 >= 0 && i0 < LEN) ? xrow[i0] : 0.0f;
        const float v1 = (i1 >= 0 && i1 < LEN) ? xrow[i1] : 0.0f;
        sVal[i] = (v0 * (1.0f - frac) + v1 * frac) * attn;
    }
    __syncthreads();

    // ---- Phase 4: main GEMM: out(256 x 16) = weight(256 x 320/group) * sVal ----
    {
        const int t   = wave;                // 16 M-tiles, one per wave
        const int co0 = t * 16;
        const int g   = co0 / CPG;           // 4 M-tiles per channel group
        v8f acc = {};
        const float* wrow = weight + (size_t)(co0 + nn) * KD + khalf * 2;
        const float* vrow = sVal + ((size_t)g * KD + khalf * 2) * LT + nn;
        __builtin_prefetch(wrow, 0, 1);
        for (int ks = 0; ks < KD / 4; ++ks) {
            v2f a = *(const v2f*)(wrow + ks * 4);
            v2f bb;
            bb.x = vrow[ks * 4 * LT];
            bb.y = vrow[ks * 4 * LT + LT];
            acc = wmma_f32_16x16x4(a, bb, acc);
        }
        #pragma unroll
        for (int r = 0; r < 8; ++r) {
            const int co = co0 + r + khalf * 8;
            out[((size_t)b * COUT + co) * LOUT + l0 + nn] = acc[r] + bias[co];
        }
    }
}

extern "C" void kernel_launch(void* const* d_in, const int* in_sizes, int n_in,
                              void* d_out, int out_size, void* d_ws, size_t ws_size,
                              hipStream_t stream) {
    (void)in_sizes; (void)n_in; (void)out_size; (void)d_ws; (void)ws_size;
    const float* x      = (const float*)d_in[0];
    const float* w_off  = (const float*)d_in[1];
    const float* b_off  = (const float*)d_in[2];
    const float* weight = (const float*)d_in[3];
    const float* bias   = (const float*)d_in[4];
    float* out = (float*)d_out;

    // Dynamic LDS > 64KB: raise the per-kernel cap (idempotent, capture-safe).
    (void)hipFuncSetAttribute((const void*)deform_conv1d_fused,
                              hipFuncAttributeMaxDynamicSharedMemorySize,
                              (int)SMEM_BYTES);

    dim3 grid(LOUT / LT, BATCH);   // 512 x 8 workgroups
    deform_conv1d_fused<<<grid, NTHREADS, SMEM_BYTES, stream>>>(
        x, w_off, b_off, weight, bias, out);
}